// ParallelTacotron2Loss_42880953483620
// MI455X (gfx1250) — compile-verified
//
#include <hip/hip_runtime.h>
#include <cstdint>

typedef __attribute__((ext_vector_type(2))) float v2f;
typedef __attribute__((ext_vector_type(8))) float v8f;

#define T_LEN 512
#define C_LEN 80
#define K_ITERS 6
#define B_BATCH 16
#define KB (K_ITERS * B_BATCH) /* 96 */
#define Z_DIM 32
#define L_TXT 128
#define GAMMA_F 0.05f
#define WARP_F 256.0f
#define BIG_F 1000000000.0f

// ---------------------------------------------------------------------------
// Phase 0: per-row squared norms for x (mel_iters rows) and y (mel rows).
// ---------------------------------------------------------------------------
__global__ void norms_kernel(const float* __restrict__ x, const float* __restrict__ y,
                             float* __restrict__ normX, float* __restrict__ normY) {
    int idx = blockIdx.x * blockDim.x + threadIdx.x;
    if (idx < KB * T_LEN) {
        const float* p = x + (size_t)idx * C_LEN;
        float s = 0.0f;
        #pragma unroll 4
        for (int c = 0; c < C_LEN; ++c) { float v = p[c]; s += v * v; }
        normX[idx] = s;
    } else {
        int r = idx - KB * T_LEN;
        if (r < B_BATCH * T_LEN) {
            const float* p = y + (size_t)r * C_LEN;
            float s = 0.0f;
            #pragma unroll 4
            for (int c = 0; c < C_LEN; ++c) { float v = p[c]; s += v * v; }
            normY[r] = s;
        }
    }
}

// ---------------------------------------------------------------------------
// Phase 1: D[b,i,j] = |x_i|^2 + |y_j|^2 - 2 x_i . y_j via f32 WMMA.
// One wave per 16x16 output tile; K-loop of 20 V_WMMA_F32_16X16X4_F32.
// A layout (16x4 f32): lane holds A[M=lane&15][K=2*(lane>>4)+v] in VGPR v.
// B layout (4x16 f32): lane holds B[K=2*(lane>>4)+v][N=lane&15] in VGPR v.
// -> both operands are one contiguous float2 load per lane per K-step.
// C/D layout: VGPR r, lane l -> (M = r + 8*(l>>4), N = l&15).
// ---------------------------------------------------------------------------
__global__ void gram_kernel(const float* __restrict__ x, const float* __restrict__ y,
                            const float* __restrict__ normX, const float* __restrict__ normY,
                            float* __restrict__ D) {
    int gwave = (blockIdx.x * blockDim.x + threadIdx.x) >> 5;
    int lane  = threadIdx.x & 31;

    int b  = gwave >> 10;        // batch 0..95 (1024 tiles per batch)
    int t2 = gwave & 1023;
    int ti = t2 >> 5;            // i-tile 0..31
    int tj = t2 & 31;            // j-tile 0..31
    int i0 = ti << 4, j0 = tj << 4;

    int rrow = lane & 15;            // matrix row within tile (A: M, B: N)
    int koff = (lane >> 4) << 1;     // K sub-offset: 0 or 2
    int yb   = b & (B_BATCH - 1);    // mel batch = b % 16 (broadcast over K)

    const float* xrow = x + ((size_t)b  * T_LEN + i0 + rrow) * C_LEN + koff;
    const float* yrow = y + ((size_t)yb * T_LEN + j0 + rrow) * C_LEN + koff;

    v8f acc = {};
    #pragma unroll
    for (int k0 = 0; k0 < C_LEN; k0 += 4) {
        v2f av = *(const v2f*)(xrow + k0);
        v2f bv = *(const v2f*)(yrow + k0);
        acc = __builtin_amdgcn_wmma_f32_16x16x4_f32(
            /*neg_a=*/false, av, /*neg_b=*/false, bv,
            /*c_mod=*/(short)0, acc, /*reuse_a=*/false, /*reuse_b=*/false);
    }

    float yn = normY[yb * T_LEN + j0 + rrow];  // column norm: fixed N per lane
    const float* nx = normX + b * T_LEN + i0;
    int half8 = (lane >> 4) << 3;
    float* Dbase = D + (size_t)b * T_LEN * T_LEN;
    #pragma unroll
    for (int r = 0; r < 8; ++r) {
        int m = r + half8;
        Dbase[(size_t)(i0 + m) * T_LEN + j0 + rrow] = nx[m] + yn - 2.0f * acc[r];
    }
}

// ---------------------------------------------------------------------------
// Phase 2: soft-DTW anti-diagonal wavefront. One workgroup per batch,
// 512 threads (16 wave32s), rows in LDS, one barrier per diagonal.
// R[i,j] = D[i,j] + softmin_g(R[i-1,j-1], R[i-1,j]+warp, R[i,j-1]+warp)
// ---------------------------------------------------------------------------
__global__ void __launch_bounds__(T_LEN)
sdtw_kernel(const float* __restrict__ D, float* __restrict__ out) {
    __shared__ float buf0[T_LEN + 1];
    __shared__ float buf1[T_LEN + 1];
    __shared__ float buf2[T_LEN + 1];
    float* r2 = buf0;   // R diagonal k-2
    float* r1 = buf1;   // R diagonal k-1
    float* rn = buf2;   // R diagonal k (being written)

    const int i = threadIdx.x;   // 0..511  (row index of the DP)
    const int b = blockIdx.x;    // batch

    r1[i] = BIG_F; r2[i] = BIG_F;
    if (i == 0) { r1[T_LEN] = BIG_F; r2[T_LEN] = BIG_F; r2[0] = 0.0f; }
    __syncthreads();

    const float* Db = D + (size_t)b * T_LEN * T_LEN + (size_t)i * T_LEN;
    const float inv_g = 1.0f / GAMMA_F;

    for (int k = 0; k < 2 * T_LEN - 1; ++k) {
        int j = k - i;
        float nv = BIG_F;
        if (j >= 0 && j < T_LEN) {
            float a  = -r2[i] * inv_g;                 // (i-1, j-1)
            float bb = -(r1[i]     + WARP_F) * inv_g;  // (i-1, j)
            float cc = -(r1[i + 1] + WARP_F) * inv_g;  // (i,   j-1)
            float m  = fmaxf(a, fmaxf(bb, cc));
            float s  = __expf(a - m) + __expf(bb - m) + __expf(cc - m);
            nv = Db[j] - GAMMA_F * (__logf(s) + m);
        }
        rn[i + 1] = nv;
        if (i == 0) rn[0] = BIG_F;
        __syncthreads();
        float* t = r2; r2 = r1; r1 = rn; rn = t;   // carry <- (r_new, r1)
    }
    if (i == 0) out[b] = r1[T_LEN];
}

// ---------------------------------------------------------------------------
// Phase 3: scalar losses + final combine (single small block).
// ---------------------------------------------------------------------------
__global__ void finalize_kernel(const float* __restrict__ sdtw,
                                const float* __restrict__ dur,
                                const float* __restrict__ mus,
                                const float* __restrict__ log_vars,
                                const int*   __restrict__ mel_length,
                                const int*   __restrict__ step_p,
                                float*       __restrict__ out) {
    __shared__ float skl[256];
    __shared__ float smel[B_BATCH];
    __shared__ float sdur[B_BATCH];
    int tid = threadIdx.x;

    // KL partials over 16*32 = 512 elements
    float kl = 0.0f;
    for (int idx = tid; idx < B_BATCH * Z_DIM; idx += 256) {
        float lv = log_vars[idx], mu = mus[idx];
        kl += 1.0f + lv - mu * mu - expf(lv);
    }
    skl[tid] = kl;

    if (tid < B_BATCH) {
        float s = 0.0f;
        for (int k = 0; k < K_ITERS; ++k) s += sdtw[k * B_BATCH + tid];
        float len = (float)mel_length[tid];
        smel[tid] = s / ((float)K_ITERS * len);
        float ds = 0.0f;
        for (int l = 0; l < L_TXT; ++l) ds += dur[tid * L_TXT + l];
        sdur[tid] = fabsf(ds - len);
    }
    __syncthreads();

    if (tid == 0) {
        float klsum = 0.0f;
        for (int i = 0; i < 256; ++i) klsum += skl[i];
        float kl_loss = -0.5f * klsum;
        float mel_loss = 0.0f, dur_loss = 0.0f;
        for (int i = 0; i < B_BATCH; ++i) { mel_loss += smel[i]; dur_loss += sdur[i]; }
        mel_loss *= (1.0f / B_BATCH);
        dur_loss  = 2.0f * dur_loss * (1.0f / B_BATCH);
        int st = step_p[0];
        float beta = (st < 3000) ? 0.0f
                   : (st >= 15000) ? 1.0f
                   : (float)(st - 3000) * (1.0f / 12000.0f);
        out[0] = mel_loss + dur_loss + beta * kl_loss;
        out[1] = mel_loss;
        out[2] = dur_loss;
        out[3] = kl_loss;
        out[4] = beta;
    }
}

// ---------------------------------------------------------------------------
extern "C" void kernel_launch(void* const* d_in, const int* in_sizes, int n_in,
                              void* d_out, int out_size, void* d_ws, size_t ws_size,
                              hipStream_t stream) {
    const float* mel_iters  = (const float*)d_in[0]; // [6,16,512,80]
    const float* mel        = (const float*)d_in[1]; // [16,512,80]
    const float* dur        = (const float*)d_in[2]; // [16,128]
    const float* mus        = (const float*)d_in[3]; // [16,32]
    const float* log_vars   = (const float*)d_in[4]; // [16,32]
    const int*   mel_length = (const int*)d_in[5];   // [16]
    const int*   step_p     = (const int*)d_in[7];   // scalar
    float* out = (float*)d_out;                      // 5 floats

    // Workspace layout (floats): D[96*512*512] | normX[96*512] | normY[16*512] | sdtw[96]
    float* ws = (float*)d_ws;
    size_t D_elems = (size_t)KB * T_LEN * T_LEN;      // 25,165,824 (~96 MB, L2-resident)
    float* D     = ws;
    float* normX = D + D_elems;
    float* normY = normX + (size_t)KB * T_LEN;
    float* sdtw  = normY + (size_t)B_BATCH * T_LEN;

    int nrows = KB * T_LEN + B_BATCH * T_LEN;         // 57344
    norms_kernel<<<(nrows + 255) / 256, 256, 0, stream>>>(mel_iters, mel, normX, normY);

    int total_waves = KB * 32 * 32;                   // 98304 tiles, 8 waves/block
    gram_kernel<<<total_waves / 8, 256, 0, stream>>>(mel_iters, mel, normX, normY, D);

    sdtw_kernel<<<KB, T_LEN, 0, stream>>>(D, sdtw);

    finalize_kernel<<<1, 256, 0, stream>>>(sdtw, dur, mus, log_vars, mel_length, step_p, out);
}